// MeanMagLayer_39290360824570
// MI455X (gfx1250) — compile-verified
//
#include <hip/hip_runtime.h>

// MeanMagLayer on MI455X (gfx1250).
//
// out[b,t] = m0[b] + S2[t+1]/(t+3),  S2 = cumsum(cumsum(e)),
// e = [data[b,0,MAG_COL], data[b,0:N,W-1]]   (length N+1)
//
// Bandwidth-bound by the strided column gather (stride 160B -> one 128B line
// per element, ~67MB @ 23.3 TB/s ~ 3us). Data movement uses the CDNA5 Tensor
// Data Mover: each wave issues one tensor_load_to_lds descriptor for a
// 256-row slice of the column (tile_dim0=1, tensor_dim0_stride=40), tracked
// by TENSORcnt. Serial scan dependence is broken with the FP32 matrix pipe:
// cumsum of 16 segments x 16 elems = L(16x16) @ V(16x16) via four chained
// V_WMMA_F32_16X16X4_F32 ops; cross-segment offsets via Hillis-Steele in LDS.

typedef __attribute__((ext_vector_type(2))) float        v2f;
typedef __attribute__((ext_vector_type(8))) float        v8f;
typedef __attribute__((ext_vector_type(4))) unsigned int v4u;
typedef __attribute__((ext_vector_type(4))) int          v4i;
typedef __attribute__((ext_vector_type(8))) int          v8i;

#define WCOLS   40
#define MAG_COL 20

#if defined(__has_builtin)
#if __has_builtin(__builtin_amdgcn_tensor_load_to_lds)
#define HAVE_TDM 1
#endif
#endif
#if __has_include(<hip/amd_detail/amd_gfx1250_TDM.h>)
#define TDM_6ARG 1   // therock-10.0 headers => clang-23 6-arg builtin
#endif

constexpr int SEG      = 16;            // elements per WMMA column (segment)
constexpr int TILE     = 256;           // 16x16 elements per WMMA tile
constexpr int NTHREADS = 1024;          // 32 wave32 waves per WGP
constexpr int NWAVES   = NTHREADS / 32;
constexpr int MAXN     = 8192;
constexpr int MP       = 8448;          // pad N+1=8193 to 33 tiles of 256
constexpr int NTILES   = MP / TILE;     // 33
constexpr int NSEG     = MP / SEG;      // 528

__global__ __launch_bounds__(NTHREADS)
void meanmag_dscan_kernel(const float* __restrict__ data,
                          const float* __restrict__ batch_m0,
                          float* __restrict__ out, int N) {
  __shared__ float X[MP];       // working column; scanned in place, twice
  __shared__ float segA[NSEG];  // segment totals / scan ping
  __shared__ float segB[NSEG];  // scan pong

  const int tid  = threadIdx.x;
  const int b    = blockIdx.x;
  const int lane = tid & 31;    // wave32
  const int wave = tid >> 5;
  const int half = lane >> 4;   // WMMA lane-half (K / row-group select)
  const int m    = lane & 15;   // WMMA row (A) / column (B, C, D)

  if (N > MAXN) return;         // uniform shape guard

  const size_t base = (size_t)b * (size_t)N * WCOLS;

  // ---- 1) bring the strided column e[0..N] into LDS, zero-padded to MP ---
  bool loaded = false;
#if defined(HAVE_TDM)
  if ((N % NWAVES) == 0 && (N / NWAVES) <= 0xFFFF) {
    // Per-wave TDM slice: rows of data[b, w*rows:(w+1)*rows, W-1] -> X[1+w*rows].
    const int      rows = N / NWAVES;                       // 256 for N=8192
    const int      w    = __builtin_amdgcn_readfirstlane((int)(tid >> 5));
    const uint64_t ga   = (uint64_t)(uintptr_t)
        (data + base + (size_t)w * rows * WCOLS + (WCOLS - 1));
    const uint32_t la   = (uint32_t)(uintptr_t)&X[1 + w * rows]; // LDS byte off

    // D# group 0: count=1 | lds_addr | global_addr | type=2 ("image")
    v4u g0;
    g0[0] = 1u;
    g0[1] = la;
    g0[2] = (uint32_t)(ga & 0xFFFFFFFFu);
    g0[3] = (uint32_t)((ga >> 32) & 0x1FFFFFFu) | (2u << 30);
    // D# group 1: data_size=2 (4B) | tensor_dim0=40 | tensor_dim1=rows |
    //             tile_dim0=1 | tile_dim1=rows | tensor_dim0_stride=40
    v8i g1;
    g1[0] = (int)(2u << 16);
    g1[1] = (int)((uint32_t)WCOLS << 16);
    g1[2] = (int)(((uint32_t)rows & 0xFFFFu) << 16);
    g1[3] = (int)((((uint32_t)rows >> 16) & 0xFFFFu) | (1u << 16));
    g1[4] = (int)((uint32_t)rows & 0xFFFFu);
    g1[5] = WCOLS;
    g1[6] = 0;
    g1[7] = 0;
    v4i g2 = {0, 0, 0, 0};      // unused for 2D tile
    v4i g3 = {0, 0, 0, 0};
#if defined(TDM_6ARG)
    v8i g4 = {0, 0, 0, 0, 0, 0, 0, 0};
    __builtin_amdgcn_tensor_load_to_lds(g0, g1, g2, g3, g4, 0);
#else
    __builtin_amdgcn_tensor_load_to_lds(g0, g1, g2, g3, 0);
#endif
    // e[0] and tail padding via normal stores while the DMA is in flight.
    if (tid == 0) X[0] = data[base + MAG_COL];
    for (int i = N + 1 + tid; i < MP; i += NTHREADS) X[i] = 0.0f;
    __builtin_amdgcn_s_wait_tensorcnt(0);
    loaded = true;
  }
#endif
  if (!loaded) {                 // vector-gather fallback path
    for (int i = tid; i < MP; i += NTHREADS) {
      float v = 0.0f;
      if (i == 0)      v = data[base + MAG_COL];
      else if (i <= N) v = data[base + (size_t)(i - 1) * WCOLS + (WCOLS - 1)];
      X[i] = v;
    }
  }
  __syncthreads();

  // ---- A operand: lower-triangular ones L(16x16), split into 4 K-chunks --
  // ISA 16x4 f32 A layout: lanes 0-15 hold K={0,1} in v[0],v[1];
  // lanes 16-31 hold K={2,3}.  Global column k = 4c + 2*half + v.
  v2f a[4];
#pragma unroll
  for (int c = 0; c < 4; ++c) {
    const int k0 = 4 * c + 2 * half;
    a[c].x = (k0     <= m) ? 1.0f : 0.0f;
    a[c].y = (k0 + 1 <= m) ? 1.0f : 0.0f;
  }

  // ---- 2) two inclusive scan passes: X <- cumsum(X), then again ----------
#pragma unroll 1
  for (int pass = 0; pass < 2; ++pass) {
    // (a) intra-segment scans: P = L @ V per 16x16 tile (4 chained WMMAs).
    //     Wave-uniform loop => EXEC all-ones at every WMMA.
    for (int tile = wave; tile < NTILES; tile += NWAVES) {
      float* Xt = &X[tile * TILE];
      const int n = m;                       // segment (column) index
      v2f bb[4];
#pragma unroll
      for (int c = 0; c < 4; ++c) {          // B 4x16: row k, col n
        const int k = 4 * c + 2 * half;
        bb[c].x = Xt[n * SEG + k];
        bb[c].y = Xt[n * SEG + k + 1];
      }
      v8f acc = {};
      acc = __builtin_amdgcn_wmma_f32_16x16x4_f32(false, a[0], false, bb[0],
                                                  (short)0, acc, false, false);
      acc = __builtin_amdgcn_wmma_f32_16x16x4_f32(false, a[1], false, bb[1],
                                                  (short)0, acc, false, false);
      acc = __builtin_amdgcn_wmma_f32_16x16x4_f32(false, a[2], false, bb[2],
                                                  (short)0, acc, false, false);
      acc = __builtin_amdgcn_wmma_f32_16x16x4_f32(false, a[3], false, bb[3],
                                                  (short)0, acc, false, false);
      // D 16x16 f32: VGPR p holds row M = p + 8*half, column n = lane%16
#pragma unroll
      for (int p = 0; p < 8; ++p)
        Xt[n * SEG + 8 * half + p] = acc[p];
      if (half)                              // row 15 = segment total
        segA[tile * (TILE / SEG) + n] = acc[7];
    }
    __syncthreads();

    // (b) Hillis-Steele inclusive scan of NSEG segment totals (10 steps)
    float* src = segA;
    float* dst = segB;
    for (int off = 1; off < NSEG; off <<= 1) {
      if (tid < NSEG) {
        float v = src[tid];
        if (tid >= off) v += src[tid - off];
        dst[tid] = v;                        // dst != src: no race with reads
      }
      __syncthreads();
      float* t = src; src = dst; dst = t;
    }

    // (c) add exclusive segment offsets -> full inclusive scan in X
    for (int i = tid; i < MP; i += NTHREADS) {
      const int s = i >> 4;
      if (s > 0) X[i] += src[s - 1];
    }
    __syncthreads();
  }

  // ---- 3) emit: out[t] = m0 + S2[t+1]/(t+3) -------------------------------
  const float  m0    = batch_m0[b];
  const size_t obase = (size_t)b * (size_t)N;
  for (int t = tid; t < N; t += NTHREADS)
    out[obase + t] = m0 + X[t + 1] / (float)(t + 3);
}

extern "C" void kernel_launch(void* const* d_in, const int* in_sizes, int n_in,
                              void* d_out, int out_size, void* d_ws, size_t ws_size,
                              hipStream_t stream) {
  const float* data = (const float*)d_in[0];
  // d_in[1] = batch_N (int32): unused by the reference computation.
  const float* m0   = (const float*)d_in[2];
  float*       out  = (float*)d_out;

  const int B = in_sizes[2];                         // batch_M0 element count
  const int N = (B > 0) ? in_sizes[0] / (B * WCOLS)  // data is (B, N, W)
                        : 0;

  if (B <= 0 || N <= 0) return;
  meanmag_dscan_kernel<<<dim3(B), dim3(NTHREADS), 0, stream>>>(data, m0, out, N);
}